// Upscale2d_28638841930310
// MI455X (gfx1250) — compile-verified
//
#include <hip/hip_runtime.h>

// Upscale2d (StyleGAN2 upsample_2d, factor=2, FIR k=[1,3,3,1]) for MI455X.
// Memory-bound op: 640MB traffic -> ~27.5us floor at 23.3 TB/s. Strategy:
// async global->LDS staging (CDNA5 ASYNCcnt path), separable polyphase
// compute in registers, nontemporal float4 coalesced stores (TH_NT) so the
// 512MB write stream doesn't evict the 128MB input from the 192MB L2.

#define H_IN   256
#define W_IN   256
#define TI     16              // input rows per block
#define TROWS  (TI + 2)        // staged rows: [i0-1, i0+TI]
#define W_OUT  (2 * W_IN)
#define H_OUT  (2 * H_IN)

#if defined(__gfx1250__) && __has_builtin(__builtin_amdgcn_global_load_async_to_lds_b128)
#define USE_ASYNC_BUILTIN 1
#endif

typedef int   vint4   __attribute__((vector_size(16)));
typedef float vfloat4 __attribute__((ext_vector_type(4)));
#if defined(__gfx1250__)
typedef __attribute__((address_space(1))) vint4* gvec4_ptr;
typedef __attribute__((address_space(3))) vint4* lvec4_ptr;
#endif

__device__ __forceinline__ void async_copy16(const float* gp, float* lp) {
#if defined(USE_ASYNC_BUILTIN)
  // (global src, lds dst, imm offset, imm cpol)
  __builtin_amdgcn_global_load_async_to_lds_b128(
      (gvec4_ptr)(void*)gp, (lvec4_ptr)(void*)lp, 0, 0);
#elif defined(__gfx1250__)
  // GV mode: vdst = LDS byte address VGPR, vaddr = 64-bit global address pair.
  unsigned lds_addr = (unsigned)(unsigned long long)lp;
  unsigned long long gaddr = (unsigned long long)gp;
  asm volatile("global_load_async_to_lds_b128 %0, %1, off"
               :: "v"(lds_addr), "v"(gaddr) : "memory");
#else
  *(vfloat4*)lp = *(const vfloat4*)gp;
#endif
}

__device__ __forceinline__ void wait_async_done() {
#if defined(__gfx1250__)
#if __has_builtin(__builtin_amdgcn_s_wait_asynccnt)
  __builtin_amdgcn_s_wait_asynccnt(0);
#else
  asm volatile("s_wait_asynccnt 0" ::: "memory");
#endif
#endif
}

__device__ __forceinline__ void store_nt16(float* p, vfloat4 v) {
#if __has_builtin(__builtin_nontemporal_store)
  __builtin_nontemporal_store(v, (vfloat4*)p);
#else
  *(vfloat4*)p = v;
#endif
}

__global__ __launch_bounds__(256)
void Upscale2d_fir_kernel(const float* __restrict__ in, float* __restrict__ out) {
  __shared__ float tile[TROWS][W_IN];   // 18 KB

  const int plane = blockIdx.x >> 4;           // H_IN/TI == 16 row-blocks per plane
  const int i0    = (blockIdx.x & 15) * TI;
  const int tid   = threadIdx.x;

  const float* pin  = in  + (size_t)plane * (H_IN * W_IN);
  float*       pout = out + (size_t)plane * ((size_t)H_OUT * W_OUT);

  // ---- Stage input rows [i0-1, i0+TI] into LDS (zero outside image) ----
  // TROWS * (W_IN/4) = 18 * 64 = 1152 sixteen-byte chunks, 256 threads.
  for (int idx = tid; idx < TROWS * (W_IN / 4); idx += 256) {
    const int tr = idx >> 6;          // W_IN/4 == 64
    const int q  = idx & 63;
    const int gr = i0 - 1 + tr;       // absolute input row
    float* lp = &tile[tr][q * 4];
    if (gr >= 0 && gr < H_IN) {
      async_copy16(pin + (size_t)gr * W_IN + q * 4, lp);
    } else {
      lp[0] = 0.f; lp[1] = 0.f; lp[2] = 0.f; lp[3] = 0.f;
    }
  }
  // Warm caches for the next row-block of this plane (speculative; harmless).
  if (i0 + TI < H_IN)
    __builtin_prefetch(pin + (size_t)(i0 + TI) * W_IN + (tid & 63) * 4, 0, 1);

  wait_async_done();
  __syncthreads();

  // ---- Compute: polyphase upsample-by-2 with k1d = [1,3,3,1]/4 per axis ----
  // thread -> 2 input cols (tx) x 8 input rows (ty half)
  const int tx = tid & 127;           // 128 col-groups * 2 cols = 256
  const int ty = tid >> 7;            // 0..1, 8 rows each
  const int c0 = tx * 2;

  #pragma unroll
  for (int rr = 0; rr < TI / 2; ++rr) {
    const int r = ty * (TI / 2) + rr;       // local input row 0..15
    const int i = i0 + r;                   // absolute input row
    // tile[r] = row i-1, tile[r+1] = row i, tile[r+2] = row i+1
    float ve[4], vo[4];
    #pragma unroll
    for (int k = 0; k < 4; ++k) {
      const int c = c0 - 1 + k;
      float a = 0.f, b = 0.f, d = 0.f;
      if (c >= 0 && c < W_IN) {
        a = tile[r][c];
        b = tile[r + 1][c];
        d = tile[r + 2][c];
      }
      ve[k] = fmaf(3.f, b, a) * 0.25f;      // even out row: (x[i-1] + 3 x[i]) / 4
      vo[k] = fmaf(3.f, b, d) * 0.25f;      // odd  out row: (3 x[i] + x[i+1]) / 4
    }
    // horizontal polyphase over the vertically-filtered values
    vfloat4 e, o;
    e.x = fmaf(3.f, ve[1], ve[0]) * 0.25f;  // out col 2c0
    e.y = fmaf(3.f, ve[1], ve[2]) * 0.25f;  // out col 2c0+1
    e.z = fmaf(3.f, ve[2], ve[1]) * 0.25f;  // out col 2c0+2
    e.w = fmaf(3.f, ve[2], ve[3]) * 0.25f;  // out col 2c0+3
    o.x = fmaf(3.f, vo[1], vo[0]) * 0.25f;
    o.y = fmaf(3.f, vo[1], vo[2]) * 0.25f;
    o.z = fmaf(3.f, vo[2], vo[1]) * 0.25f;
    o.w = fmaf(3.f, vo[2], vo[3]) * 0.25f;

    store_nt16(pout + (size_t)(2 * i)     * W_OUT + 2 * c0, e);
    store_nt16(pout + (size_t)(2 * i + 1) * W_OUT + 2 * c0, o);
  }
}

extern "C" void kernel_launch(void* const* d_in, const int* in_sizes, int n_in,
                              void* d_out, int out_size, void* d_ws, size_t ws_size,
                              hipStream_t stream) {
  const float* x = (const float*)d_in[0];
  float* y = (float*)d_out;
  const int planes = in_sizes[0] / (H_IN * W_IN);       // 8*64 = 512
  dim3 grid(planes * (H_IN / TI));                      // 8192 blocks
  Upscale2d_fir_kernel<<<grid, 256, 0, stream>>>(x, y);
}